// GaussianKernel_31138512896695
// MI455X (gfx1250) — compile-verified
//
#include <hip/hip_runtime.h>

typedef float v2f __attribute__((ext_vector_type(2)));
typedef float v8f __attribute__((ext_vector_type(8)));

#define C_DIM 128
#define M_DIM 512
#define ROW_TILES 2304          // 36864 / 16
#define LDS_STRIDE 132          // 128 + 4 pad words, avoids LDS bank conflicts

// ---------------------------------------------------------------------------
// Precompute w2[m] = sum_c w[c][m]^2  (block-invariant, done once into d_ws)
// ---------------------------------------------------------------------------
__global__ void __launch_bounds__(256) w2_kernel(const float* __restrict__ w,
                                                 float* __restrict__ w2) {
    int m = blockIdx.x * 256 + threadIdx.x;
    if (m < M_DIM) {
        float s = 0.f;
        #pragma unroll 8
        for (int c = 0; c < C_DIM; ++c) {
            float v = w[c * M_DIM + m];
            s = fmaf(v, v, s);
        }
        w2[m] = s;
    }
}

// ---------------------------------------------------------------------------
// Main kernel: one block per 16-row tile, 8 waves x 64 columns each.
// out[n][m] = exp(-(x2[n] + w2[m] - 2 * (x . w))) + b[m]
// ---------------------------------------------------------------------------
__device__ __forceinline__ void epilogue_tile(const v8f& acc, float wb, float bb,
                                              const float* __restrict__ x2r,
                                              float* __restrict__ outp) {
    #pragma unroll
    for (int v = 0; v < 8; ++v) {
        float d2 = x2r[v] + wb - 2.0f * acc[v];
        outp[(size_t)v * M_DIM] = __expf(-d2) + bb;
    }
}

__global__ void __launch_bounds__(256, 4) gauss_kernel(
    const float* __restrict__ x, const float* __restrict__ w,
    const float* __restrict__ b, const float* __restrict__ w2,
    float* __restrict__ out) {
    __shared__ __align__(16) float xs[16 * LDS_STRIDE];
    __shared__ float x2s[16];

    const int tid  = threadIdx.x;
    const int lane = tid & 31;
    const int wave = tid >> 5;
    const int row0 = blockIdx.x * 16;

    // ---- cooperative load: 16x128 f32 x-tile -> LDS (512 float4, 2/thread)
    const float4* xg = (const float4*)(x + (size_t)row0 * C_DIM);
    #pragma unroll
    for (int i = 0; i < 2; ++i) {
        int f  = tid + i * 256;       // flat float4 index in tile
        int r  = f >> 5;              // row (0..15)
        int k4 = f & 31;              // float4 within row
        float4 v = xg[r * 32 + k4];
        *(float4*)&xs[r * LDS_STRIDE + k4 * 4] = v;
    }
    __syncthreads();

    // ---- row norms ||x_n||^2 (wave 0, lanes 0..15), consumed after 2nd barrier
    if (wave == 0 && lane < 16) {
        const float* rp = &xs[lane * LDS_STRIDE];
        float s = 0.f;
        #pragma unroll 8
        for (int k = 0; k < C_DIM; ++k) s = fmaf(rp[k], rp[k], s);
        x2s[lane] = s;
    }

    const int kh = lane >> 4;     // K half-select (A layout: lanes 16-31 hold K+2)
    const int ln = lane & 15;     // row (A) / column (B,C,D) within tile
    const int m0 = wave * 64;     // this wave's first output column

    v8f acc0 = {}, acc1 = {}, acc2 = {}, acc3 = {};

    // A fragment source: lane ln holds row ln; VGPR j holds K = k0 + 2*kh + j
    const float* arow = &xs[ln * LDS_STRIDE + 2 * kh];

    for (int k0 = 0; k0 < C_DIM; k0 += 4) {
        v2f a;
        a.x = arow[k0];
        a.y = arow[k0 + 1];

        // B fragment: lane ln holds column m0+ln(+16t); VGPR j holds K = k0+2*kh+j
        const float* wp = w + (size_t)(k0 + 2 * kh) * M_DIM + m0 + ln;
        v2f b0, b1, b2, b3;
        b0.x = wp[0];   b0.y = wp[M_DIM];
        b1.x = wp[16];  b1.y = wp[M_DIM + 16];
        b2.x = wp[32];  b2.y = wp[M_DIM + 32];
        b3.x = wp[48];  b3.y = wp[M_DIM + 48];

        acc0 = __builtin_amdgcn_wmma_f32_16x16x4_f32(false, a, false, b0, (short)0, acc0, false, false);
        acc1 = __builtin_amdgcn_wmma_f32_16x16x4_f32(false, a, false, b1, (short)0, acc1, false, false);
        acc2 = __builtin_amdgcn_wmma_f32_16x16x4_f32(false, a, false, b2, (short)0, acc2, false, false);
        acc3 = __builtin_amdgcn_wmma_f32_16x16x4_f32(false, a, false, b3, (short)0, acc3, false, false);
    }
    __syncthreads();

    // ---- epilogue: C/D layout: VGPR v, lane L -> row 8*kh + v, col ln
    float x2r[8];
    #pragma unroll
    for (int v = 0; v < 8; ++v) x2r[v] = x2s[kh * 8 + v];

    float* outp = out + (size_t)(row0 + kh * 8) * M_DIM + m0 + ln;
    {
        int m = m0 + ln;
        epilogue_tile(acc0, w2[m],      b[m],      x2r, outp);
        epilogue_tile(acc1, w2[m + 16], b[m + 16], x2r, outp + 16);
        epilogue_tile(acc2, w2[m + 32], b[m + 32], x2r, outp + 32);
        epilogue_tile(acc3, w2[m + 48], b[m + 48], x2r, outp + 48);
    }
}

extern "C" void kernel_launch(void* const* d_in, const int* in_sizes, int n_in,
                              void* d_out, int out_size, void* d_ws, size_t ws_size,
                              hipStream_t stream) {
    const float* x = (const float*)d_in[0];
    const float* w = (const float*)d_in[1];
    const float* b = (const float*)d_in[2];
    float* out = (float*)d_out;
    float* w2  = (float*)d_ws;   // 512 floats of scratch

    w2_kernel<<<(M_DIM + 255) / 256, 256, 0, stream>>>(w, w2);
    gauss_kernel<<<ROW_TILES, 256, 0, stream>>>(x, w, b, w2, out);
}